// GAT_34806414967126
// MI455X (gfx1250) — compile-verified
//
#include <hip/hip_runtime.h>

// ---------------------------------------------------------------------------
// Types for WMMA
// ---------------------------------------------------------------------------
typedef __attribute__((ext_vector_type(2))) float v2f;
typedef __attribute__((ext_vector_type(8))) float v8f;

#define NNODES 100000
#define NEDGES 1200000
#define ENC_NEG_INF 0x007FFFFFu   // enc(-inf) under the ordered-uint mapping

// Ordered-uint mapping: preserves float total order under unsigned compare.
__device__ __forceinline__ unsigned enc_f(float f) {
    unsigned u = __float_as_uint(f);
    return (u & 0x80000000u) ? ~u : (u | 0x80000000u);
}
__device__ __forceinline__ float dec_f(unsigned u) {
    return __uint_as_float((u & 0x80000000u) ? (u & 0x7fffffffu) : ~u);
}

__device__ __forceinline__ void atomicAddF(float* p, float v) {
    // Lowers to global_atomic_add_f32 on gfx1250 (no CAS loop).
    unsafeAtomicAdd(p, v);
}

// ---------------------------------------------------------------------------
// Generic fills (workspace must be re-initialized every call)
// ---------------------------------------------------------------------------
__global__ void fill_u32(unsigned* __restrict__ p, unsigned v, long n) {
    long i = (long)blockIdx.x * blockDim.x + threadIdx.x;
    if (i < n) p[i] = v;
}
__global__ void fill_f32(float* __restrict__ p, float v, long n) {
    long i = (long)blockIdx.x * blockDim.x + threadIdx.x;
    if (i < n) p[i] = v;
}

// ---------------------------------------------------------------------------
// Weight packing: fold attention vectors into the projection so one GEMM
// produces xs, a_s, a_d and the skip projection simultaneously.
// Layer1 Bcat: [64 x 336]: [0,256)=W1_src  [256,260)=W1_src.att_src
//                          [260,264)=W1_dst.att_dst  [264,328)=Wl1  [328,336)=0
// ---------------------------------------------------------------------------
__global__ void pack_b1(const float* __restrict__ W_src, const float* __restrict__ W_dst,
                        const float* __restrict__ att_src, const float* __restrict__ att_dst,
                        const float* __restrict__ Wl, float* __restrict__ Bcat) {
    int k = blockIdx.x;      // 0..63 input feature
    int t = threadIdx.x;     // 0..335 output column
    float v = 0.0f;
    if (t < 256) {
        v = W_src[k * 256 + t];
    } else if (t < 260) {
        int h = t - 256; float s = 0.0f;
        for (int c = 0; c < 64; ++c) s += W_src[k * 256 + h * 64 + c] * att_src[h * 64 + c];
        v = s;
    } else if (t < 264) {
        int h = t - 260; float s = 0.0f;
        for (int c = 0; c < 64; ++c) s += W_dst[k * 256 + h * 64 + c] * att_dst[h * 64 + c];
        v = s;
    } else if (t < 328) {
        v = Wl[k * 64 + (t - 264)];
    }
    Bcat[k * 336 + t] = v;
}

// Layer2 Bcat: [64 x 32]: [0,8)=W2_src  [8,12)=W2_src.att_src
//                         [12,16)=W2_dst.att_dst  [16,18)=Wl2  [18,32)=0
__global__ void pack_b2(const float* __restrict__ W_src, const float* __restrict__ W_dst,
                        const float* __restrict__ att_src, const float* __restrict__ att_dst,
                        const float* __restrict__ Wl, float* __restrict__ Bcat) {
    int k = blockIdx.x;      // 0..63
    int t = threadIdx.x;     // 0..31
    float v = 0.0f;
    if (t < 8) {
        v = W_src[k * 8 + t];
    } else if (t < 12) {
        int h = t - 8; float s = 0.0f;
        for (int c = 0; c < 2; ++c) s += W_src[k * 8 + h * 2 + c] * att_src[h * 2 + c];
        v = s;
    } else if (t < 16) {
        int h = t - 12; float s = 0.0f;
        for (int c = 0; c < 2; ++c) s += W_dst[k * 8 + h * 2 + c] * att_dst[h * 2 + c];
        v = s;
    } else if (t < 18) {
        v = Wl[k * 2 + (t - 16)];
    }
    Bcat[k * 32 + t] = v;
}

// ---------------------------------------------------------------------------
// WMMA f32 GEMM: C[M,NB] = A[M,64] @ B[64,NB], row-major, NB % 16 == 0.
// Block = 256 threads = 8 waves; each wave one 16x16 tile (8 M-tiles/block).
// The block's 64x16 B panel (4 KB) is staged in LDS once and reused by all
// 8 waves -> inner loop is 1x global_load_b64 (A pair) + 2x ds_load_b32 (B),
// no 64-bit address arithmetic. K=64 -> 16 chained v_wmma_f32_16x16x4_f32.
//
// ISA layouts (wave32):
//   A 16x4:  lanes 0-15 m=lane, vgpr{0,1}=K{0,1}; lanes 16-31 vgpr{0,1}=K{2,3}
//   B 4x16:  lanes 0-15 n=lane, vgpr{0,1}=K{0,1}; lanes 16-31 vgpr{0,1}=K{2,3}
//   C 16x16: vgpr i: M=i (lanes 0-15) / M=i+8 (lanes 16-31), N=lane%16
// LDS bank check: dword addr (k0+kb)*16+ln -> halves {16k0+ln} / {16k0+32+ln}
// are disjoint mod 64 banks: conflict-free.
// ---------------------------------------------------------------------------
__global__ void __launch_bounds__(256) gemm_k64_wmma(
        const float* __restrict__ A, const float* __restrict__ B,
        float* __restrict__ C, int M, int NB) {
    __shared__ float Bs[64 * 16];

    const int col0 = blockIdx.x * 16;
    // Cooperative stage of the 64x16 B panel into LDS (4 elems/thread).
#pragma unroll
    for (int i = threadIdx.x; i < 64 * 16; i += 256) {
        Bs[i] = B[(size_t)(i >> 4) * NB + col0 + (i & 15)];
    }
    __syncthreads();

    const int wave = threadIdx.x >> 5;
    const int lane = threadIdx.x & 31;
    const int row0 = (blockIdx.y * 8 + wave) * 16;
    if (row0 >= M) return;

    const int ln  = lane & 15;          // row of A tile / col of B tile
    const int grp = lane >> 4;          // half-wave selects K pair
    const int kb  = grp * 2;

    v8f acc = {};
    const float* Arow = A + (size_t)(row0 + ln) * 64 + kb;

#pragma unroll
    for (int k0 = 0; k0 < 64; k0 += 4) {
        v2f a = *(const v2f*)(Arow + k0);           // global_load_b64, imm offset
        v2f b;
        b.x = Bs[(k0 + kb) * 16 + ln];              // ds_load, imm offsets
        b.y = Bs[(k0 + kb + 1) * 16 + ln];
        acc = __builtin_amdgcn_wmma_f32_16x16x4_f32(
            /*neg_a=*/false, a, /*neg_b=*/false, b,
            /*c_mod=*/(short)0, acc, /*reuse_a=*/false, /*reuse_b=*/false);
    }

    float* Crow = C + (size_t)row0 * NB + col0 + ln;
#pragma unroll
    for (int i = 0; i < 8; ++i) {
        Crow[(size_t)(i + 8 * grp) * NB] = acc[i];
    }
}

// ---------------------------------------------------------------------------
// Edge phase 1: e = leaky_relu(a_s[src] + a_d[dst]); segment-max into m_enc.
// a_s / a_d live at column offsets inside the fused GEMM output Y.
// ---------------------------------------------------------------------------
__global__ void edge_lrelu_max(const int* __restrict__ src, const int* __restrict__ dst,
                               const float* __restrict__ Y, int ldy, int as_off, int ad_off,
                               float* __restrict__ e_out, unsigned* __restrict__ m_enc,
                               int nE) {
    int e = blockIdx.x * blockDim.x + threadIdx.x;
    if (e >= nE) return;
    int s = src[e], d = dst[e];
#pragma unroll
    for (int h = 0; h < 4; ++h) {
        float v = Y[(size_t)s * ldy + as_off + h] + Y[(size_t)d * ldy + ad_off + h];
        v = (v > 0.0f) ? v : 0.2f * v;
        e_out[(size_t)e * 4 + h] = v;
        atomicMax(&m_enc[(size_t)d * 4 + h], enc_f(v));   // global_atomic_max_u32
    }
}

// Decode segment max (nodes with no in-edge -> 0, like the reference) and zero s.
__global__ void decode_max_init_s(const unsigned* __restrict__ m_enc,
                                  float* __restrict__ m, float* __restrict__ s, int n) {
    int i = blockIdx.x * blockDim.x + threadIdx.x;
    if (i >= n) return;
    float v = dec_f(m_enc[i]);
    if (v < -3.0e38f) v = 0.0f;     // only -inf can appear as "empty segment"
    m[i] = v;
    s[i] = 0.0f;
}

// Edge phase 2: p = exp(e - m[dst]) (stored back in place); s[dst] += p.
__global__ void edge_exp_sum(const int* __restrict__ dst, const float* __restrict__ m,
                             float* __restrict__ ep, float* __restrict__ s, int nE) {
    int e = blockIdx.x * blockDim.x + threadIdx.x;
    if (e >= nE) return;
    int d = dst[e];
#pragma unroll
    for (int h = 0; h < 4; ++h) {
        float p = __expf(ep[(size_t)e * 4 + h] - m[(size_t)d * 4 + h]);
        ep[(size_t)e * 4 + h] = p;
        atomicAddF(&s[(size_t)d * 4 + h], p);
    }
}

// Edge phase 2b: alpha = p / (s[dst] + eps), in place over p. Keeps the random
// s-gather and the divide out of the 64-threads-per-edge scatter kernel.
__global__ void edge_alpha(const int* __restrict__ dst, const float* __restrict__ s,
                           float* __restrict__ pa, int nE) {
    int e = blockIdx.x * blockDim.x + threadIdx.x;
    if (e >= nE) return;
    int d = dst[e];
#pragma unroll
    for (int h = 0; h < 4; ++h) {
        pa[(size_t)e * 4 + h] = pa[(size_t)e * 4 + h] / (s[(size_t)d * 4 + h] + 1e-16f);
    }
}

// Edge phase 3 (layer 1, C=64): 64 threads per edge, float4 gathers + fp32 atomics.
__global__ void edge_scatter64(const int* __restrict__ src, const int* __restrict__ dst,
                               const float* __restrict__ alpha,
                               const float* __restrict__ XS, int ldx,
                               float* __restrict__ agg, int nE) {
    long t = (long)blockIdx.x * blockDim.x + threadIdx.x;
    int e = (int)(t >> 6);
    if (e >= nE) return;
    int l  = (int)(t & 63);
    int h  = l >> 4;
    int c0 = (l & 15) * 4;
    int sn = src[e], dn = dst[e];
    float al = alpha[(size_t)e * 4 + h];
    const float4 v = *(const float4*)&XS[(size_t)sn * ldx + h * 64 + c0];
    float* o = &agg[(size_t)dn * 256 + h * 64 + c0];
    atomicAddF(o + 0, v.x * al);
    atomicAddF(o + 1, v.y * al);
    atomicAddF(o + 2, v.z * al);
    atomicAddF(o + 3, v.w * al);
}

// Edge phase 3 (layer 2, C=2): one thread per edge handles all H*C = 8 values.
__global__ void edge_scatter2(const int* __restrict__ src, const int* __restrict__ dst,
                              const float* __restrict__ alpha,
                              const float* __restrict__ XS, int ldx,
                              float* __restrict__ agg, int nE) {
    int e = blockIdx.x * blockDim.x + threadIdx.x;
    if (e >= nE) return;
    int sn = src[e], dn = dst[e];
#pragma unroll
    for (int h = 0; h < 4; ++h) {
        float al = alpha[(size_t)e * 4 + h];
        float v0 = XS[(size_t)sn * ldx + h * 2 + 0];
        float v1 = XS[(size_t)sn * ldx + h * 2 + 1];
        atomicAddF(&agg[(size_t)dn * 8 + h * 2 + 0], v0 * al);
        atomicAddF(&agg[(size_t)dn * 8 + h * 2 + 1], v1 * al);
    }
}

// Layer1 epilogue: head-mean + GAT bias + linear skip (from fused GEMM) + relu.
__global__ void finalize1(const float* __restrict__ agg, const float* __restrict__ Y,
                          int ldy, int skip_off, const float* __restrict__ b1,
                          const float* __restrict__ bl1, float* __restrict__ hout, int n) {
    long t = (long)blockIdx.x * blockDim.x + threadIdx.x;
    if (t >= (long)n * 64) return;
    int node = (int)(t >> 6), c = (int)(t & 63);
    float a = 0.0f;
#pragma unroll
    for (int h = 0; h < 4; ++h) a += agg[(size_t)node * 256 + h * 64 + c];
    float v = a * 0.25f + b1[c] + Y[(size_t)node * ldy + skip_off + c] + bl1[c];
    hout[t] = (v > 0.0f) ? v : 0.0f;
}

// Layer2 epilogue: head-mean + bias + skip + sigmoid -> d_out [N,2].
__global__ void finalize2(const float* __restrict__ agg, const float* __restrict__ Y,
                          int ldy, int skip_off, const float* __restrict__ b2,
                          const float* __restrict__ bl2, float* __restrict__ out, int n) {
    long t = (long)blockIdx.x * blockDim.x + threadIdx.x;
    if (t >= (long)n * 2) return;
    int node = (int)(t >> 1), c = (int)(t & 1);
    float a = 0.0f;
#pragma unroll
    for (int h = 0; h < 4; ++h) a += agg[(size_t)node * 8 + h * 2 + c];
    float v = a * 0.25f + b2[c] + Y[(size_t)node * ldy + skip_off + c] + bl2[c];
    out[t] = 1.0f / (1.0f + __expf(-v));
}

// ---------------------------------------------------------------------------
// Host launcher
// ---------------------------------------------------------------------------
extern "C" void kernel_launch(void* const* d_in, const int* in_sizes, int n_in,
                              void* d_out, int out_size, void* d_ws, size_t ws_size,
                              hipStream_t stream) {
    const float* x        = (const float*)d_in[0];
    const int*   eidx     = (const int*)  d_in[1];
    const float* W1_src   = (const float*)d_in[2];
    const float* W1_dst   = (const float*)d_in[3];
    const float* att1_src = (const float*)d_in[4];
    const float* att1_dst = (const float*)d_in[5];
    const float* b1       = (const float*)d_in[6];
    const float* Wl1      = (const float*)d_in[7];
    const float* bl1      = (const float*)d_in[8];
    const float* W2_src   = (const float*)d_in[9];
    const float* W2_dst   = (const float*)d_in[10];
    const float* att2_src = (const float*)d_in[11];
    const float* att2_dst = (const float*)d_in[12];
    const float* b2       = (const float*)d_in[13];
    const float* Wl2      = (const float*)d_in[14];
    const float* bl2      = (const float*)d_in[15];

    const int N = NNODES, E = NEDGES;
    const int* src = eidx;          // edge_index[0]
    const int* dst = eidx + E;      // edge_index[1]

    // ---- carve workspace ----
    char* w = (char*)d_ws;
    auto carve = [&](size_t bytes) -> void* {
        void* p = (void*)w;
        w += (bytes + 255) & ~(size_t)255;
        return p;
    };
    float*    Bcat1 = (float*)   carve((size_t)64 * 336 * 4);
    float*    Bcat2 = (float*)   carve((size_t)64 * 32 * 4);
    float*    Y1    = (float*)   carve((size_t)N * 336 * 4);   // xs | a_s | a_d | skip
    float*    Y2    = (float*)   carve((size_t)N * 32 * 4);
    unsigned* m_enc = (unsigned*)carve((size_t)N * 4 * 4);
    float*    m     = (float*)   carve((size_t)N * 4 * 4);
    float*    ssum  = (float*)   carve((size_t)N * 4 * 4);
    float*    ebuf  = (float*)   carve((size_t)E * 4 * 4);     // e -> p -> alpha
    float*    agg1  = (float*)   carve((size_t)N * 256 * 4);
    float*    agg2  = (float*)   carve((size_t)N * 8 * 4);
    float*    hbuf  = (float*)   carve((size_t)N * 64 * 4);
    (void)ws_size; (void)n_in; (void)in_sizes; (void)out_size;

    const int TB = 256;
    auto blocks = [](long n, int tb) { return (unsigned)((n + tb - 1) / tb); };
    const unsigned gemmY = (unsigned)(((N / 16) + 7) / 8);     // 6250 M-tiles / 8 waves

    // ================= Layer 1 =================
    pack_b1<<<64, 336, 0, stream>>>(W1_src, W1_dst, att1_src, att1_dst, Wl1, Bcat1);
    gemm_k64_wmma<<<dim3(336 / 16, gemmY), TB, 0, stream>>>(x, Bcat1, Y1, N, 336);

    fill_u32<<<blocks((long)N * 4, TB), TB, 0, stream>>>(m_enc, ENC_NEG_INF, (long)N * 4);
    edge_lrelu_max<<<blocks(E, TB), TB, 0, stream>>>(src, dst, Y1, 336, 256, 260,
                                                     ebuf, m_enc, E);
    decode_max_init_s<<<blocks((long)N * 4, TB), TB, 0, stream>>>(m_enc, m, ssum, N * 4);
    edge_exp_sum<<<blocks(E, TB), TB, 0, stream>>>(dst, m, ebuf, ssum, E);
    edge_alpha<<<blocks(E, TB), TB, 0, stream>>>(dst, ssum, ebuf, E);

    fill_f32<<<blocks((long)N * 256, TB), TB, 0, stream>>>(agg1, 0.0f, (long)N * 256);
    edge_scatter64<<<blocks((long)E * 64, TB), TB, 0, stream>>>(src, dst, ebuf,
                                                                Y1, 336, agg1, E);
    finalize1<<<blocks((long)N * 64, TB), TB, 0, stream>>>(agg1, Y1, 336, 264,
                                                           b1, bl1, hbuf, N);

    // ================= Layer 2 =================
    pack_b2<<<64, 32, 0, stream>>>(W2_src, W2_dst, att2_src, att2_dst, Wl2, Bcat2);
    gemm_k64_wmma<<<dim3(32 / 16, gemmY), TB, 0, stream>>>(hbuf, Bcat2, Y2, N, 32);

    fill_u32<<<blocks((long)N * 4, TB), TB, 0, stream>>>(m_enc, ENC_NEG_INF, (long)N * 4);
    edge_lrelu_max<<<blocks(E, TB), TB, 0, stream>>>(src, dst, Y2, 32, 8, 12,
                                                     ebuf, m_enc, E);
    decode_max_init_s<<<blocks((long)N * 4, TB), TB, 0, stream>>>(m_enc, m, ssum, N * 4);
    edge_exp_sum<<<blocks(E, TB), TB, 0, stream>>>(dst, m, ebuf, ssum, E);
    edge_alpha<<<blocks(E, TB), TB, 0, stream>>>(dst, ssum, ebuf, E);

    fill_f32<<<blocks((long)N * 8, TB), TB, 0, stream>>>(agg2, 0.0f, (long)N * 8);
    edge_scatter2<<<blocks(E, TB), TB, 0, stream>>>(src, dst, ebuf, Y2, 32, agg2, E);
    finalize2<<<blocks((long)N * 2, TB), TB, 0, stream>>>(agg2, Y2, 32, 16,
                                                          b2, bl2, (float*)d_out, N);
}